// TokenAdaptiveNorm_12352325943309
// MI455X (gfx1250) — compile-verified
//
#include <hip/hip_runtime.h>
#include <math.h>

// ---------------- problem constants (from reference) ----------------
constexpr int B_   = 32;
constexpr int L_   = 8192;
constexpr int C_   = 32;
constexpr int P_   = 16;
constexpr int S_   = 8;
constexpr int N_   = (L_ - P_) / S_ + 1;   // 1023 tokens
constexpr int K_   = 16;
constexpr int HID_ = 32;
constexpr int NSEG = L_ / S_;              // 1024 eight-row segments

typedef __attribute__((ext_vector_type(2))) float v2f;
typedef __attribute__((ext_vector_type(8))) float v8f;

// =====================================================================
// Kernel 1: per-(b, seg, c) sums of x and x^2 over 8 rows.
// Patch n stats = segment n + segment n+1 (patches overlap by S=8).
// Layout of output: (b, seg, c) so the normalize pass reads coalesced.
// =====================================================================
__global__ void seg_stats_kernel(const float* __restrict__ x,
                                 float* __restrict__ segSum,
                                 float* __restrict__ segSq) {
    const int tid = threadIdx.x;          // 256 threads = (c:32, r:8)
    const int c = tid & 31;
    const int r = tid >> 5;
    const int b = blockIdx.x >> 10;       // grid = B * NSEG
    const int m = blockIdx.x & (NSEG - 1);

    const size_t xi = (((size_t)b * L_) + (size_t)m * 8 + r) * C_ + c;
    const float v = x[xi];

    __shared__ float ss[256];
    __shared__ float sq[256];
    ss[tid] = v;
    sq[tid] = v * v;
    __syncthreads();

    if (r == 0) {
        float s = 0.f, q = 0.f;
#pragma unroll
        for (int i = 0; i < 8; ++i) { s += ss[i * 32 + c]; q += sq[i * 32 + c]; }
        const size_t o = ((size_t)b * NSEG + m) * C_ + c;
        segSum[o] = s;
        segSq[o]  = q;
    }
}

// =====================================================================
// Kernel 2: global per-(b,c) mean / 1/std from the segment sums.
// =====================================================================
__global__ void global_stats_kernel(const float* __restrict__ segSum,
                                    const float* __restrict__ segSq,
                                    float* __restrict__ muG,
                                    float* __restrict__ rstdG) {
    const int tid = threadIdx.x;          // 256 threads = (c:32, r:8)
    const int c = tid & 31;
    const int r = tid >> 5;
    const int b = blockIdx.x;             // grid = B

    float s = 0.f, q = 0.f;
    for (int m = r; m < NSEG; m += 8) {
        const size_t o = ((size_t)b * NSEG + m) * C_ + c;
        s += segSum[o];
        q += segSq[o];
    }
    __shared__ float ss[256];
    __shared__ float sq[256];
    ss[tid] = s;
    sq[tid] = q;
    __syncthreads();

    if (r == 0) {
        float S = 0.f, Q = 0.f;
#pragma unroll
        for (int i = 0; i < 8; ++i) { S += ss[i * 32 + c]; Q += sq[i * 32 + c]; }
        const float mu  = S * (1.0f / (float)L_);
        const float var = Q * (1.0f / (float)L_) - mu * mu;
        const float std = sqrtf(fmaxf(var, 1e-4f));   // STD_FLOOR^2
        muG[b * C_ + c]   = mu;
        rstdG[b * C_ + c] = 1.0f / std;               // std >= 0.01 already
    }
}

// =====================================================================
// Kernel 3: alpha gate MLP via V_WMMA_F32_16X16X4_F32.
//   per ff row (b*C+c): (N x 16) @ (16 x 32) + b1 -> gelu -> @ (32 x 1) -> sigmoid
// One wave per 16-token tile. K=16 processed as 4 chained K=4 WMMAs,
// HID=32 as two N=16 output tiles. Bias b1 pre-loaded into accumulators.
// beta branch of the reference is dead code and skipped.
// Output alpha stored in (b, n, c) layout for coalesced reads later.
// =====================================================================
__global__ void alpha_mlp_wmma_kernel(const float* __restrict__ ff,
                                      const float* __restrict__ w1,  // (16, 32)
                                      const float* __restrict__ b1,  // (32,)
                                      const float* __restrict__ w2,  // (32,)
                                      const float* __restrict__ b2,  // (1,)
                                      float* __restrict__ alphaOut) {
    const int lane = threadIdx.x & 31;
    const int wave = (int)blockIdx.x * (blockDim.x >> 5) + (threadIdx.x >> 5);

    const int row  = wave >> 6;   // 0..1023 == b*C + c
    const int tile = wave & 63;   // 64 tiles of 16 tokens (last partial)

    const int col  = lane & 15;          // N-column within 16-wide tile
    const int koff = (lane >> 4) * 2;    // K sub-pair per ISA A/B layout

    // ---- A fragments: 16 tokens x K=16, four K=4 slices (v2f per lane) ----
    const int   m0   = tile * 16 + col;  // token index this lane's A row
    const bool  mok  = (m0 < N_);
    const float* arow = ff + (size_t)row * ((size_t)N_ * K_) + (size_t)m0 * K_;
    v2f a[4];
#pragma unroll
    for (int q = 0; q < 4; ++q) {
        if (mok) a[q] = *(const v2f*)(arow + q * 4 + koff);
        else     { a[q][0] = 0.f; a[q][1] = 0.f; }
    }

    // ---- B fragments: w1 (K=16 x HID=32) -> two 16-col tiles x 4 K-slices ----
    v2f bA[4], bB[4];
#pragma unroll
    for (int q = 0; q < 4; ++q) {
        const int kb = q * 4 + koff;
        bA[q][0] = w1[(kb + 0) * HID_ + col];
        bA[q][1] = w1[(kb + 1) * HID_ + col];
        bB[q][0] = w1[(kb + 0) * HID_ + 16 + col];
        bB[q][1] = w1[(kb + 1) * HID_ + 16 + col];
    }

    // ---- accumulators start at bias b1 (per output column) ----
    const float bias0 = b1[col];
    const float bias1 = b1[16 + col];
    v8f acc0, acc1;
#pragma unroll
    for (int j = 0; j < 8; ++j) { acc0[j] = bias0; acc1[j] = bias1; }

    // ---- chained WMMAs: D = A*B + C, f32 16x16x4 ----
#pragma unroll
    for (int q = 0; q < 4; ++q) {
        acc0 = __builtin_amdgcn_wmma_f32_16x16x4_f32(
                   false, a[q], false, bA[q], (short)0, acc0, false, false);
        acc1 = __builtin_amdgcn_wmma_f32_16x16x4_f32(
                   false, a[q], false, bB[q], (short)0, acc1, false, false);
    }

    // ---- epilogue: exact gelu, dot with w2, sigmoid ----
    // C/D layout: VGPR j, lanes 0-15 -> (row j,   col lane)
    //                     lanes 16-31 -> (row j+8, col lane-16)
    const float w2a = w2[col];
    const float w2b = w2[16 + col];
    const float bias2 = b2[0];

    float p[8];
#pragma unroll
    for (int j = 0; j < 8; ++j) {
        const float h0 = acc0[j];
        const float h1 = acc1[j];
        const float g0 = 0.5f * h0 * (1.0f + erff(h0 * 0.7071067811865476f));
        const float g1 = 0.5f * h1 * (1.0f + erff(h1 * 0.7071067811865476f));
        p[j] = g0 * w2a + g1 * w2b;
    }
    // reduce across the 16 lanes of each half (rows live per-half)
#pragma unroll
    for (int j = 0; j < 8; ++j) {
        p[j] += __shfl_xor(p[j], 1, 32);
        p[j] += __shfl_xor(p[j], 2, 32);
        p[j] += __shfl_xor(p[j], 4, 32);
        p[j] += __shfl_xor(p[j], 8, 32);
    }

    if (col == 0) {                      // lanes 0 and 16 write 8 rows each
        const int rbase = (lane >> 4) * 8;
        const int b = row >> 5;
        const int c = row & 31;
#pragma unroll
        for (int j = 0; j < 8; ++j) {
            const int m = tile * 16 + rbase + j;
            if (m < N_) {
                const float sig = 1.0f / (1.0f + expf(-(p[j] + bias2)));
                alphaOut[((size_t)b * N_ + m) * C_ + c] = sig;
            }
        }
    }
}

// =====================================================================
// Kernel 4: streaming normalize. Per element (b, t, c):
//   n(t) = nearest token (argmin ties -> lower index), local stats from
//   segments n, n+1, blend with global stats by alpha, affine.
// =====================================================================
__global__ void norm_kernel(const float* __restrict__ x,
                            const float* __restrict__ segSum,
                            const float* __restrict__ segSq,
                            const float* __restrict__ muG,
                            const float* __restrict__ rstdG,
                            const float* __restrict__ alpha,
                            const float* __restrict__ scale,
                            const float* __restrict__ bias,
                            float* __restrict__ out) {
    const size_t idx = (size_t)blockIdx.x * blockDim.x + threadIdx.x;
    const int c = (int)(idx & 31);
    const int t = (int)((idx >> 5) & (L_ - 1));
    const int b = (int)(idx >> 18);

    // tok_per_time: centers at 8n+8; tie (dist equal) picks smaller n.
    const int u = t - 8;
    int n = 0;
    if (u >= 0) {
        n = (u >> 3) + (((u & 7) > 4) ? 1 : 0);
        if (n > N_ - 1) n = N_ - 1;
    }

    const size_t sb = ((size_t)b * NSEG + n) * C_ + c;
    const float s = segSum[sb] + segSum[sb + C_];
    const float q = segSq[sb]  + segSq[sb + C_];
    const float mu_l  = s * (1.0f / 16.0f);
    const float var_l = q * (1.0f / 16.0f) - mu_l * mu_l;
    const float std_l = sqrtf(fmaxf(var_l, 1e-4f));   // >= 0.01 -> is the denom

    const float xv = x[idx];
    float xl = (xv - mu_l) / std_l;
    xl = fminf(30.0f, fmaxf(-30.0f, xl));

    const float mg = muG[b * C_ + c];
    float xg = (xv - mg) * rstdG[b * C_ + c];
    xg = fminf(30.0f, fmaxf(-30.0f, xg));

    const float a = alpha[((size_t)b * N_ + n) * C_ + c];
    out[idx] = (a * xl + (1.0f - a) * xg) * scale[c] + bias[c];
}

// =====================================================================
// launch
// =====================================================================
extern "C" void kernel_launch(void* const* d_in, const int* in_sizes, int n_in,
                              void* d_out, int out_size, void* d_ws, size_t ws_size,
                              hipStream_t stream) {
    const float* x     = (const float*)d_in[0];   // (B, L, C)
    const float* ff    = (const float*)d_in[1];   // (B*C, N, K)
    const float* aw1   = (const float*)d_in[2];   // (K, HID)
    const float* ab1   = (const float*)d_in[3];   // (HID,)
    const float* aw2   = (const float*)d_in[4];   // (HID, 1)
    const float* ab2   = (const float*)d_in[5];   // (1,)
    // d_in[6..9] = beta MLP params: dead code in the reference, unused.
    const float* ascl  = (const float*)d_in[10];  // (C,)
    const float* abias = (const float*)d_in[11];  // (C,)
    float* out = (float*)d_out;

    // workspace layout (floats); total ~12.6 MB
    float* wsf    = (float*)d_ws;
    float* segSum = wsf;                                   // B*NSEG*C = 1048576
    float* segSq  = segSum + (size_t)B_ * NSEG * C_;       // 1048576
    float* muG    = segSq  + (size_t)B_ * NSEG * C_;       // B*C = 1024
    float* rstdG  = muG + B_ * C_;                         // 1024
    float* alphaW = rstdG + B_ * C_;                       // B*N*C = 1047552

    // 1) segment sums (reads x once)
    seg_stats_kernel<<<B_ * NSEG, 256, 0, stream>>>(x, segSum, segSq);

    // 2) global stats from segment sums
    global_stats_kernel<<<B_, 256, 0, stream>>>(segSum, segSq, muG, rstdG);

    // 3) alpha gate MLP (WMMA f32 16x16x4): B*C rows * 64 tiles, 8 waves/block
    {
        const int totalWaves = B_ * C_ * ((N_ + 15) / 16);  // 65536
        const int blocks = totalWaves / 8;                  // 8192
        alpha_mlp_wmma_kernel<<<blocks, 256, 0, stream>>>(ff, aw1, ab1, aw2, ab2, alphaW);
    }

    // 4) streaming normalize (reads x again, writes out)
    {
        const size_t total = (size_t)B_ * L_ * C_;          // 8388608
        norm_kernel<<<(int)(total / 256), 256, 0, stream>>>(
            x, segSum, segSq, muG, rstdG, alphaW, ascl, abias, out);
    }
}